// QuantumHead_33148557591178
// MI455X (gfx1250) — compile-verified
//
#include <hip/hip_runtime.h>
#include <math.h>

typedef float v2f __attribute__((ext_vector_type(2)));
typedef float v8f __attribute__((ext_vector_type(8)));

#define NQ    12
#define NLAY  8
#define DIM   4096   // 2^12 amplitudes
#define INSZ  96     // NLAY * NQ

// D(16x16,f32) = A(16x4,f32) x B(4x16,f32) + C   -- wave32 CDNA5 WMMA
static __device__ __forceinline__ v8f wmma_f32_16x16x4(v2f a, v2f b, v8f c) {
  return __builtin_amdgcn_wmma_f32_16x16x4_f32(false, a, false, b, (short)0, c,
                                               false, false);
}

// Apply fused 16x16 (complex) unitary of wire-group G to the LDS state.
// State index i = (i0<<8)|(i1<<4)|i2 ; group G acts on nibble i_G.
// One call = one 16x16x16 complex matmul tile T; tiles are region-local
// (output region == input region) so no barrier is needed inside a stage.
template<int G, bool UREAL>
static __device__ __forceinline__ void apply_group(
    int T,
    float* __restrict__ sre, float* __restrict__ sim,
    const float* __restrict__ ur, const float* __restrict__ ui,
    int lh, int hi)
{
  v8f Dr = {0.f,0.f,0.f,0.f,0.f,0.f,0.f,0.f};
  v8f Di = {0.f,0.f,0.f,0.f,0.f,0.f,0.f,0.f};

  if constexpr (G == 0) {
    // new[m,x] = sum_k U[m,k] * S[(k<<8)|x], tile = 16 columns x=16T..16T+15
    #pragma unroll
    for (int kk = 0; kk < 4; ++kk) {
      const int k0 = 4*kk + 2*hi;
      v2f Ar, Br, Bi;
      Ar.x = ur[lh*16 + k0];  Ar.y = ur[lh*16 + k0 + 1];
      const int ib0 = (k0 << 8) | (T*16 + lh);
      Br.x = sre[ib0];  Br.y = sre[ib0 + 256];
      Bi.x = sim[ib0];  Bi.y = sim[ib0 + 256];
      Dr = wmma_f32_16x16x4(Ar, Br, Dr);
      Di = wmma_f32_16x16x4(Ar, Bi, Di);
      if constexpr (!UREAL) {
        v2f Ai, nAi;
        Ai.x = ui[lh*16 + k0];  Ai.y = ui[lh*16 + k0 + 1];
        nAi.x = -Ai.x;          nAi.y = -Ai.y;
        Dr = wmma_f32_16x16x4(nAi, Bi, Dr);   // -Ui*Si
        Di = wmma_f32_16x16x4(Ai,  Br, Di);   //  Ui*Sr
      }
    }
    #pragma unroll
    for (int v = 0; v < 8; ++v) {
      const int id = ((v + 8*hi) << 8) | (T*16 + lh);
      sre[id] = Dr[v];  sim[id] = Di[v];
    }
  } else if constexpr (G == 1) {
    // tile = fixed top nibble a=T : new[a,m,c] = sum_k U[m,k] S[a,k,c]
    #pragma unroll
    for (int kk = 0; kk < 4; ++kk) {
      const int k0 = 4*kk + 2*hi;
      v2f Ar, Br, Bi;
      Ar.x = ur[lh*16 + k0];  Ar.y = ur[lh*16 + k0 + 1];
      const int ib0 = (T << 8) | (k0 << 4) | lh;
      Br.x = sre[ib0];  Br.y = sre[ib0 + 16];
      Bi.x = sim[ib0];  Bi.y = sim[ib0 + 16];
      Dr = wmma_f32_16x16x4(Ar, Br, Dr);
      Di = wmma_f32_16x16x4(Ar, Bi, Di);
      if constexpr (!UREAL) {
        v2f Ai, nAi;
        Ai.x = ui[lh*16 + k0];  Ai.y = ui[lh*16 + k0 + 1];
        nAi.x = -Ai.x;          nAi.y = -Ai.y;
        Dr = wmma_f32_16x16x4(nAi, Bi, Dr);
        Di = wmma_f32_16x16x4(Ai,  Br, Di);
      }
    }
    #pragma unroll
    for (int v = 0; v < 8; ++v) {
      const int id = (T << 8) | ((v + 8*hi) << 4) | lh;
      sre[id] = Dr[v];  sim[id] = Di[v];
    }
  } else {
    // low nibble: new[r,c] = sum_k S[r,k]*U[c,k]  (S(256x16) x U^T), tile = 16 rows
    #pragma unroll
    for (int kk = 0; kk < 4; ++kk) {
      const int k0 = 4*kk + 2*hi;
      v2f Ar, Ai, Br;
      const int ia0 = ((T*16 + lh) << 4) | k0;
      Ar.x = sre[ia0];  Ar.y = sre[ia0 + 1];
      Ai.x = sim[ia0];  Ai.y = sim[ia0 + 1];
      Br.x = ur[lh*16 + k0];  Br.y = ur[lh*16 + k0 + 1];  // B[k][n] = U[n][k]
      Dr = wmma_f32_16x16x4(Ar, Br, Dr);
      Di = wmma_f32_16x16x4(Ai, Br, Di);
      if constexpr (!UREAL) {
        v2f Bi, nBi;
        Bi.x = ui[lh*16 + k0];  Bi.y = ui[lh*16 + k0 + 1];
        nBi.x = -Bi.x;          nBi.y = -Bi.y;
        Dr = wmma_f32_16x16x4(Ai, nBi, Dr);   // -Si*Ui^T
        Di = wmma_f32_16x16x4(Ar,  Bi, Di);   //  Sr*Ui^T
      }
    }
    #pragma unroll
    for (int v = 0; v < 8; ++v) {
      const int id = ((T*16 + v + 8*hi) << 4) | lh;
      sre[id] = Dr[v];  sim[id] = Di[v];
    }
  }
}

template<bool UREAL>
static __device__ __forceinline__ void apply_all_groups(
    float* __restrict__ sre, float* __restrict__ sim,
    const float (*__restrict__ Ur)[256], const float (*__restrict__ Ui)[256],
    int wave, int lh, int hi)
{
  apply_group<0, UREAL>(2*wave + 0, sre, sim, Ur[0], Ui[0], lh, hi);
  apply_group<0, UREAL>(2*wave + 1, sre, sim, Ur[0], Ui[0], lh, hi);
  __syncthreads();
  apply_group<1, UREAL>(2*wave + 0, sre, sim, Ur[1], Ui[1], lh, hi);
  apply_group<1, UREAL>(2*wave + 1, sre, sim, Ur[1], Ui[1], lh, hi);
  __syncthreads();
  apply_group<2, UREAL>(2*wave + 0, sre, sim, Ur[2], Ui[2], lh, hi);
  apply_group<2, UREAL>(2*wave + 1, sre, sim, Ur[2], Ui[2], lh, hi);
  __syncthreads();
}

__global__ __launch_bounds__(256)
void qsim_drc12_kernel(const float* __restrict__ x,
                       const float* __restrict__ ip,
                       const float* __restrict__ wt,
                       float* __restrict__ out)
{
  const int b    = blockIdx.x;
  const int tid  = threadIdx.x;
  const int lane = tid & 31;
  const int wave = tid >> 5;
  const int lh   = lane & 15;
  const int hi   = lane >> 4;

  __shared__ float sre[DIM];        // state real
  __shared__ float sim_[DIM];       // state imag
  __shared__ float Ur[3][256];      // fused 16x16 unitary per 4-wire group
  __shared__ float Ui[3][256];
  __shared__ float w2r[12][4];      // per-wire 2x2, row-major (re)
  __shared__ float w2i[12][4];      // per-wire 2x2 (im)

  __builtin_prefetch(x + (size_t)b * INSZ, 0, 1);   // global_prefetch_b8
  __builtin_prefetch(ip, 0, 1);
  __builtin_prefetch(wt, 0, 1);

  // CNOT-ring permutation targets: layer-invariant, compute once.
  // Sequential chain (w=0..11: bit(11-t) ^= bit(11-c)) closes to:
  //   new bit p (p<=10) = XOR of original bits p..11   (suffix parity)
  //   new bit 11        = old bit 11 ^ parity(bits 0..11)
  int ji[16];
  #pragma unroll
  for (int t = 0; t < 16; ++t) {
    const unsigned i = (unsigned)(tid + (t << 8));
    unsigned s = i;
    s ^= s >> 1;  s ^= s >> 2;  s ^= s >> 4;  s ^= s >> 8;
    ji[t] = (int)((s & 0x7FFu) | ((((s ^ (i >> 11)) & 1u)) << 11));
  }

  // |0...0>
  #pragma unroll
  for (int t = 0; t < 16; ++t) {
    const int i = tid + (t << 8);
    sre[i]  = (i == 0) ? 1.0f : 0.0f;
    sim_[i] = 0.0f;
  }
  __syncthreads();

  for (int l = 0; l < NLAY; ++l) {
    for (int stage = 0; stage < 2; ++stage) {
      // ---- per-wire 2x2 matrices for this stage ----
      if (tid < NQ) {
        if (stage == 0) {
          // RY(theta), theta = input_params[l,w] * x[b,l,w]   (real matrix)
          const float ang = ip[l*NQ + tid] * x[(size_t)b*INSZ + l*NQ + tid];
          float s, c;  sincosf(0.5f * ang, &s, &c);
          w2r[tid][0] =  c;  w2r[tid][1] = -s;
          w2r[tid][2] =  s;  w2r[tid][3] =  c;
          w2i[tid][0] = 0.f; w2i[tid][1] = 0.f;
          w2i[tid][2] = 0.f; w2i[tid][3] = 0.f;
        } else {
          // Rot(phi,theta,omega) = RZ(omega) RY(theta) RZ(phi)
          const float* wp = wt + (size_t)(l*NQ + tid) * 3;
          const float phi = wp[0], th = wp[1], om = wp[2];
          float st, ct;          sincosf(0.5f * th, &st, &ct);
          float sap, cap;        sincosf(0.5f * (phi + om), &sap, &cap);
          float sam, cam;        sincosf(0.5f * (phi - om), &sam, &cam);
          w2r[tid][0] =  ct*cap;  w2i[tid][0] = -ct*sap;   // ep*c
          w2r[tid][1] = -st*cam;  w2i[tid][1] = -st*sam;   // -conj(em)*s
          w2r[tid][2] =  st*cam;  w2i[tid][2] = -st*sam;   // em*s
          w2r[tid][3] =  ct*cap;  w2i[tid][3] =  ct*sap;   // conj(ep)*c
        }
      }
      __syncthreads();

      // ---- fused 16x16 unitary per group: Kron of four 2x2 ----
      {
        const int m = tid >> 4, k = tid & 15;
        #pragma unroll
        for (int g = 0; g < 3; ++g) {
          float pr = 1.0f, pi = 0.0f;
          #pragma unroll
          for (int q = 0; q < 4; ++q) {
            const int w  = 4*g + q;
            const int e  = (((m >> (3 - q)) & 1) << 1) | ((k >> (3 - q)) & 1);
            const float er = w2r[w][e], ei = w2i[w][e];
            const float nr = pr*er - pi*ei;
            const float ni = pr*ei + pi*er;
            pr = nr;  pi = ni;
          }
          Ur[g][tid] = pr;
          Ui[g][tid] = pi;
        }
      }
      __syncthreads();

      // ---- apply the three group unitaries via WMMA (2 tiles / wave) ----
      if (stage == 0)
        apply_all_groups<true >(sre, sim_, Ur, Ui, wave, lh, hi);
      else
        apply_all_groups<false>(sre, sim_, Ur, Ui, wave, lh, hi);
    }

    // ---- CNOT ring: pure basis-state permutation, one scatter pass ----
    {
      float vr[16], vi[16];
      #pragma unroll
      for (int t = 0; t < 16; ++t) {
        const int i = tid + (t << 8);
        vr[t] = sre[i];  vi[t] = sim_[i];
      }
      __syncthreads();
      #pragma unroll
      for (int t = 0; t < 16; ++t) { sre[ji[t]] = vr[t];  sim_[ji[t]] = vi[t]; }
      __syncthreads();
    }
  }

  // ---- write statevector: complex64 = interleaved float pairs ----
  float2* ob = (float2*)(out + (size_t)b * (DIM * 2));
  #pragma unroll
  for (int t = 0; t < 16; ++t) {
    const int i = tid + (t << 8);
    ob[i] = make_float2(sre[i], sim_[i]);
  }
}

extern "C" void kernel_launch(void* const* d_in, const int* in_sizes, int n_in,
                              void* d_out, int out_size, void* d_ws, size_t ws_size,
                              hipStream_t stream) {
  (void)n_in; (void)out_size; (void)d_ws; (void)ws_size;
  const float* x  = (const float*)d_in[0];   // (B, 96)
  const float* ip = (const float*)d_in[1];   // (8, 12)
  const float* wt = (const float*)d_in[2];   // (8, 12, 3)
  float* out = (float*)d_out;                // (B, 4096) complex64 -> float pairs
  const int B = in_sizes[0] / INSZ;
  qsim_drc12_kernel<<<B, 256, 0, stream>>>(x, ip, wt, out);
}